// RotatedROIAlign_9577777070797
// MI455X (gfx1250) — compile-verified
//
#include <hip/hip_runtime.h>
#include <hip/hip_bf16.h>
#include <stdint.h>

// Problem constants (match reference)
#define B_      4
#define C_      256
#define H_      256
#define W_      256
#define NBOX    64
#define S_      7
#define SCALE   0.25f
#define NROI    (B_ * NBOX)        // 256 rois
#define P_      (S_ * S_)          // 49 sample points
#define CHUNKS  8                  // channel chunks per roi
#define CPB     (C_ / CHUNKS)      // 32 channels per block
#define WORK    (CPB * P_)         // 1568 outputs per block
#define TPB     256                // 8 wave32s
#define NITER   ((WORK + TPB - 1) / TPB)  // 7
#define DEPTH   4                  // pipeline stages (async-gather lookahead)
#define PLANE_BYTES (H_ * W_ * 4)  // 256 KB per channel plane

// ---- CDNA5 async path helpers (ASYNCcnt-tracked DMA gathers into LDS) ----

template <int N>
__device__ __forceinline__ void wait_asynccnt() {
    // S_WAIT_ASYNCCNT: wait until <= N async LDS transfers outstanding.
    asm volatile("s_wait_asynccnt %0" ::"n"(N) : "memory");
}

__device__ __forceinline__ void async_gather_b32(uint32_t lds_off, uint32_t voff,
                                                 const float* sbase) {
    // GLOBAL_LOAD_ASYNC_TO_LDS_B32 (GVS mode):
    //   copies 4 bytes from global address sbase (SGPR) + voff (VGPR) to the LDS address in lds_off
    // Fire-and-forget per-lane gather; no destination VGPR, tracked by ASYNCcnt.
    asm volatile("global_load_async_to_lds_b32 %0, %1, %2"
                 ::"v"(lds_off), "v"(voff), "s"(sbase)
                 : "memory");
}

__global__ __launch_bounds__(TPB) void rroi_align_async_kernel(
    const float* __restrict__ feats,   // (B, C, H, W)
    const float* __restrict__ rois,    // (B, NBOX, 5) = cx,cy,w,h,theta (world units)
    float* __restrict__ out)           // (B*NBOX, C, S, S)
{
    __shared__ float  stage[DEPTH][TPB * 4];  // 4-deep gather stage (16 KB)
    __shared__ int4   s_off4[P_];             // per-point 4 corner BYTE offsets
    __shared__ float4 s_w4[P_];               // per-point 4 corner weights

    const int tid   = threadIdx.x;
    const int r     = blockIdx.x;             // roi id in [0, 256)
    const int chunk = blockIdx.y;             // channel chunk in [0, CHUNKS)
    const int b     = r >> 6;                 // batch = r / NBOX

    // ---- Per-roi sample-point precompute (49 lanes, once per block) ----
    if (tid < P_) {
        const float* rr = rois + (size_t)r * 5;
        const float cx = rr[0] * SCALE, cy = rr[1] * SCALE;
        const float bw = rr[2] * SCALE, bh = rr[3] * SCALE;
        const float th = rr[4];

        const int   sy = tid / S_, sx = tid - sy * S_;
        const float lu = ((float)sx + 0.5f) / (float)S_ - 0.5f;
        const float lv = ((float)sy + 0.5f) / (float)S_ - 0.5f;
        const float u  = lu * bw, v = lv * bh;
        const float ct = cosf(th), st = sinf(th);
        const float x  = cx + u * ct - v * st;
        const float y  = cy + u * st + v * ct;

        const float x0f = floorf(x), y0f = floorf(y);
        const float x1f = x0f + 1.0f, y1f = y0f + 1.0f;
        const float wx1 = x - x0f, wx0 = 1.0f - wx1;
        const float wy1 = y - y0f, wy0 = 1.0f - wy1;

        const float vx0 = (x0f >= 0.0f && x0f <= (float)(W_ - 1)) ? 1.0f : 0.0f;
        const float vx1 = (x1f >= 0.0f && x1f <= (float)(W_ - 1)) ? 1.0f : 0.0f;
        const float vy0 = (y0f >= 0.0f && y0f <= (float)(H_ - 1)) ? 1.0f : 0.0f;
        const float vy1 = (y1f >= 0.0f && y1f <= (float)(H_ - 1)) ? 1.0f : 0.0f;

        const int xi0 = (int)fminf(fmaxf(x0f, 0.0f), (float)(W_ - 1));
        const int xi1 = (int)fminf(fmaxf(x1f, 0.0f), (float)(W_ - 1));
        const int yi0 = (int)fminf(fmaxf(y0f, 0.0f), (float)(H_ - 1));
        const int yi1 = (int)fminf(fmaxf(y1f, 0.0f), (float)(H_ - 1));

        // Corner byte offsets within a channel plane.
        s_off4[tid] = make_int4((yi0 * W_ + xi0) * 4,
                                (yi0 * W_ + xi1) * 4,
                                (yi1 * W_ + xi0) * 4,
                                (yi1 * W_ + xi1) * 4);
        s_w4[tid] = make_float4(wx0 * wy0 * vx0 * vy0,
                                wx1 * wy0 * vx1 * vy0,
                                wx0 * wy1 * vx0 * vy1,
                                wx1 * wy1 * vx1 * vy1);
    }
    __syncthreads();

    // Per-thread LDS stage slot byte offsets (generic LDS pointers carry the
    // LDS byte offset in their low 32 bits).
    uint32_t lds_slot[DEPTH];
#pragma unroll
    for (int s = 0; s < DEPTH; ++s)
        lds_slot[s] = (uint32_t)(uintptr_t)&stage[s][tid * 4];

    // Uniform SGPR base for this block's channel slab.
    const float* fb = feats + (size_t)(b * C_ + chunk * CPB) * (H_ * W_);

    auto issue = [&](int i) {
        int idx = i * TPB + tid;
        idx = (idx < WORK) ? idx : (WORK - 1);    // clamp tail (store predicated later)
        const int  cl    = idx / P_;
        const int  p     = idx - cl * P_;
        const int4 off   = s_off4[p];             // one ds_load_b128
        const uint32_t cb = (uint32_t)cl * PLANE_BYTES;
        const uint32_t l  = lds_slot[i & (DEPTH - 1)];
        async_gather_b32(l + 0,  cb + (uint32_t)off.x, fb);
        async_gather_b32(l + 4,  cb + (uint32_t)off.y, fb);
        async_gather_b32(l + 8,  cb + (uint32_t)off.z, fb);
        async_gather_b32(l + 12, cb + (uint32_t)off.w, fb);
    };

    // Prologue: fill DEPTH-1 pipeline stages.
#pragma unroll
    for (int j = 0; j < DEPTH - 1; ++j)
        issue(j);                                  // NITER (7) >= DEPTH-1 (3)

#pragma unroll 1
    for (int i = 0; i < NITER; ++i) {
        if (i + DEPTH - 1 < NITER) {
            issue(i + DEPTH - 1);
            wait_asynccnt<4 * (DEPTH - 1)>();      // oldest stage (iteration i) landed
        } else {
            wait_asynccnt<0>();                    // tail: full drain (no new issues)
        }

        const int idx  = i * TPB + tid;
        const int idxe = (idx < WORK) ? idx : (WORK - 1);
        const int cl   = idxe / P_;
        const int p    = idxe - cl * P_;

        const float4 vv = *reinterpret_cast<const float4*>(&stage[i & (DEPTH - 1)][tid * 4]);
        const float4 ww = s_w4[p];                 // one ds_load_b128
        const float  res = vv.x * ww.x + vv.y * ww.y + vv.z * ww.z + vv.w * ww.w;

        if (idx < WORK) {
            // out[(r, chunk*CPB + cl, p)] — fully coalesced (flat idx order)
            out[((size_t)r * C_ + chunk * CPB) * P_ + idx] = res;
        }
    }
}

extern "C" void kernel_launch(void* const* d_in, const int* in_sizes, int n_in,
                              void* d_out, int out_size, void* d_ws, size_t ws_size,
                              hipStream_t stream) {
    const float* feats = (const float*)d_in[0];   // B*C*H*W fp32
    const float* rois  = (const float*)d_in[1];   // B*NBOX*5 fp32
    float*       out   = (float*)d_out;           // B*NBOX*C*S*S fp32

    dim3 grid(NROI, CHUNKS);
    rroi_align_async_kernel<<<grid, TPB, 0, stream>>>(feats, rois, out);
}